// STC_16750372454848
// MI455X (gfx1250) — compile-verified
//
#include <hip/hip_runtime.h>
#include <math.h>

#define T_ 2000
#define B_ 32
#define C_ 128
#define L_ 100
#define NSTATES 302   // 3*L+2
#define NEG_INF_F (-1e30f)

typedef float v2f __attribute__((ext_vector_type(2)));
typedef float v8f __attribute__((ext_vector_type(8)));

// ---------------------------------------------------------------------------
// Kernel A: per-(t,b) logsumexp over classes 1..127 via accumulated
// V_WMMA_F32_16X16X4_F32.  One wave handles a 16-row x 128-col tile.
// All 32 b64 loads are issued first (64 VGPRs of data), so memory latency is
// paid once instead of per-chunk; then the exp + WMMA chain runs back-to-back.
// lp = log_softmax <= 0, so exp() cannot overflow and the direct sum-then-log
// needs no max-shift; __expf (~1e-7 rel err) is ample for the lse.
// A-matrix 16x4 f32 layout (ISA 7.12.2): lanes 0-15 hold K=0 (v0) / K=1 (v1),
// lanes 16-31 hold K=2 (v0) / K=3 (v1), M = lane&15.
// ---------------------------------------------------------------------------
__global__ void stc_lse_wmma(const float* __restrict__ lp,
                             float* __restrict__ lseOut) {
  const int wavesPerBlock = blockDim.x >> 5;
  const int wave = blockIdx.x * wavesPerBlock + (threadIdx.x >> 5);
  const int lane = threadIdx.x & 31;
  const int r0   = wave * 16;                 // 16 rows per tile; T_*B_/16 = 4000 tiles
  const int row  = r0 + (lane & 15);
  const int cbase = (lane < 16) ? 0 : 2;      // K split across half-waves
  const float* base = lp + (size_t)row * C_ + cbase;

  v2f raw[32];
#pragma unroll
  for (int q = 0; q < 32; ++q) {
    raw[q].x = base[4 * q + 0];
    raw[q].y = base[4 * q + 1];
  }

  v8f acc = {};
  v2f ones; ones.x = 1.0f; ones.y = 1.0f;     // B = 4x16 all-ones -> D[i][j] = sum_k A[i][k]

#pragma unroll
  for (int q = 0; q < 32; ++q) {              // 32 chunks of K=4 cover c = 0..127
    v2f a;
    a.x = __expf(raw[q].x);
    a.y = __expf(raw[q].y);
    if (q == 0 && lane < 16) a.x = 0.0f;      // exclude class 0 from the lse
    acc = __builtin_amdgcn_wmma_f32_16x16x4_f32(
        /*neg_a=*/false, a, /*neg_b=*/false, ones,
        /*c_mod=*/(short)0, acc, /*reuse_a=*/false, /*reuse_b=*/false);
  }

  // D layout: lanes 0-15 hold rows 0..7 in v0..v7; lanes 16-31 hold rows 8..15.
  if (lane == 0) {
#pragma unroll
    for (int v = 0; v < 8; ++v) lseOut[r0 + v] = logf(acc[v]);
  } else if (lane == 16) {
#pragma unroll
    for (int v = 0; v < 8; ++v) lseOut[r0 + 8 + v] = logf(acc[v]);
  }
}

// ---------------------------------------------------------------------------
// Kernel B: forward scan, one block (320 threads) per batch element.
// Alpha double-buffered in LDS; arc CSR (<=3 incoming arcs/state) built once
// per block by thread 0.  Emission slots: 0..99 = lp[tgt[j]], 100..199 =
// neglse(tgt[j]), 200 = blank lp[0], 201 = lse.
// The per-step state update is the latency-critical serial chain, so it uses
// fast __expf/__logf (args are <= 0 resp. z in [~1,3]); neglse keeps precise
// expf/log1pf because 1e-7 - exp(v-lse) can be near-cancelling.
// ---------------------------------------------------------------------------
__global__ void __launch_bounds__(320)
stc_scan(const float* __restrict__ lp,        // (T,B,C)
         const int* __restrict__ targets,     // (B,L)
         const float* __restrict__ lseBuf,    // (T*B)
         float* __restrict__ scoreOut) {      // (B)
  const int b   = blockIdx.x;
  const int tid = threadIdx.x;

  __shared__ float alphaA[NSTATES];
  __shared__ float alphaB[NSTATES];
  __shared__ float emit[202];
  __shared__ int   tgt[L_];
  __shared__ short aSrc[NSTATES][3];
  __shared__ short aSlot[NSTATES][3];
  __shared__ float aW[NSTATES][3];
  __shared__ int   cnt[NSTATES];

  if (tid < L_) tgt[tid] = targets[b * L_ + tid];
  if (tid < NSTATES) { cnt[tid] = 0; alphaA[tid] = NEG_INF_F; }
  if (tid == 0) alphaA[0] = 0.0f;
  __syncthreads();

  if (tid == 0) {
    // logwt = log(WLAST + (W0-WLAST)*exp(-NSTEP*ln2/THALF)), double to match host cast
    const float lw = (float)log(0.1 + 0.9 * exp(-M_LN2 * 1.0 / 10000.0));
    auto addArc = [&](int s, int d, int lt, int lpos, int wt, int wpos) {
      int k = cnt[d]++;
      aSrc[d][k] = (short)s;
      aSlot[d][k] = (lt == 0) ? (short)200
                  : (lt == 1) ? (short)lpos
                  : (lt == 2) ? (short)(100 + lpos)
                              : (short)201;
      float w;
      if (wt == 0)      w = 0.0f;
      else if (wt == 1) w = lw;
      else {
        int p = wpos - 1; if (p < 0) p = 0;
        w = (tgt[wpos] != tgt[p]) ? 0.0f : NEG_INF_F;
      }
      aW[d][k] = w;
    };
    const int S = 2 * L_ + 1;                  // 201
    for (int l = 0; l < S; ++l) {
      int idx = (l - 1) / 2;                   // only used when l is odd (idx >= 0)
      if ((l & 1) == 0) addArc(l, l, 0, 0, 0, 0);
      if (l > 0) {
        if (l & 1) addArc(l - 1, l, 1, idx, 0, 0);
        else       addArc(l - 1, l, 0, 0, 0, 0);
      }
      if ((l & 1) && l > 1) addArc(l - 2, l, 1, idx, 2, idx);
    }
    for (int l = 0; l <= L_; ++l) {
      int p1 = 2 * l - 1, p2 = 2 * l, c1 = S + l;
      int lt  = (l == L_) ? 3 : 2;
      int lpp = (l == L_) ? 0 : l;
      if (p1 >= 0) addArc(p1, c1, lt, lpp, 1, 0);
      addArc(p2, c1, lt, lpp, 1, 0);
      addArc(c1, c1, lt, lpp, 1, 0);
      if (l < L_) addArc(c1, 2 * l + 1, 1, l, 0, 0);
      addArc(c1, p2, 0, 0, 0, 0);
    }
  }
  __syncthreads();

  // Pad states with < 3 incoming arcs (weight -1e30 => contributes ~0).
  if (tid < NSTATES) {
    for (int k = cnt[tid]; k < 3; ++k) {
      aSrc[tid][k] = 0; aSlot[tid][k] = 200; aW[tid][k] = NEG_INF_F;
    }
  }
  __syncthreads();

  float* cur = alphaA;
  float* nxt = alphaB;

  for (int t = 0; t < T_; ++t) {
    const float* row = lp + ((size_t)t * B_ + b) * C_;
    const float  ls  = lseBuf[t * B_ + b];
    if (tid < L_) {
      float v = row[tgt[tid]];
      emit[tid]        = v;                                      // lp[tgt]
      emit[100 + tid]  = ls + log1pf(1e-7f - expf(v - ls));      // neglse(tgt)
    } else if (tid == 100) {
      emit[200] = row[0];                                        // blank
    } else if (tid == 101) {
      emit[201] = ls;                                            // star (lse)
    } else if (tid == 102 && t + 1 < T_) {
      __builtin_prefetch(lp + ((size_t)(t + 1) * B_ + b) * C_, 0, 1); // global_prefetch_b8
    }
    __syncthreads();

    if (tid < NSTATES) {
      float sc0 = cur[aSrc[tid][0]] + aW[tid][0] + emit[aSlot[tid][0]];
      float sc1 = cur[aSrc[tid][1]] + aW[tid][1] + emit[aSlot[tid][1]];
      float sc2 = cur[aSrc[tid][2]] + aW[tid][2] + emit[aSlot[tid][2]];
      float m = fmaxf(fmaxf(sc0, sc1), fmaxf(sc2, NEG_INF_F));
      float z = __expf(sc0 - m) + __expf(sc1 - m) + __expf(sc2 - m);
      nxt[tid] = m + __logf(z);
    }
    __syncthreads();
    float* tmp = cur; cur = nxt; nxt = tmp;
  }

  if (tid == 0) {
    // ACCEPT = {2L-1, 2L, 3L+1} = {199, 200, 301}
    float a0 = cur[199], a1 = cur[200], a2 = cur[301];
    float m = fmaxf(a0, fmaxf(a1, a2));
    scoreOut[b] = m + logf(expf(a0 - m) + expf(a1 - m) + expf(a2 - m));
  }
}

// ---------------------------------------------------------------------------
// Kernel C: loss = mean_b(-score[b] / T)
// ---------------------------------------------------------------------------
__global__ void stc_finalize(const float* __restrict__ score,
                             float* __restrict__ out) {
  if (threadIdx.x == 0 && blockIdx.x == 0) {
    float acc = 0.0f;
    for (int i = 0; i < B_; ++i) acc += -score[i] / (float)T_;
    out[0] = acc / (float)B_;
  }
}

extern "C" void kernel_launch(void* const* d_in, const int* in_sizes, int n_in,
                              void* d_out, int out_size, void* d_ws, size_t ws_size,
                              hipStream_t stream) {
  const float* inputs  = (const float*)d_in[0];   // (T,B,C) f32 log-softmax
  const int*   targets = (const int*)d_in[1];     // (B,L) i32
  float* out = (float*)d_out;                     // scalar f32 loss

  float* lseBuf   = (float*)d_ws;                 // T*B floats
  float* scoreBuf = lseBuf + (size_t)T_ * B_;     // B floats

  // Phase 1: 64000 rows / 16 rows per wave = 4000 waves = 500 blocks x 8 waves
  stc_lse_wmma<<<500, 256, 0, stream>>>(inputs, lseBuf);
  // Phase 2: one 320-thread block per batch element
  stc_scan<<<B_, 320, 0, stream>>>(inputs, targets, lseBuf, scoreBuf);
  // Phase 3: scalar reduction
  stc_finalize<<<1, 32, 0, stream>>>(scoreBuf, out);
}